// PPM_35390530519157
// MI455X (gfx1250) — compile-verified
//
#include <hip/hip_runtime.h>
#include <hip/hip_bf16.h>

// ---------------------------------------------------------------------------
// Fused cosine-similarity feature aggregation for MI455X (gfx1250, wave32).
//
//   dots  = X^T X          (per batch, K=256)         -> V_WMMA_F32_16X16X4_F32
//   sim   = relu(dots/(n_p n_q))^2                    -> fused, never hits HBM
//   tx    = W X + bias                                -> V_WMMA_F32_16X16X4_F32
//   out   = tx . sim^T     (K=4096)                   -> V_WMMA_F32_16X16X4_F32
//
// Workspace: tx (16.8 MB) + norms (64 KB). sim (268 MB) is regenerated in
// registers/LDS tile-by-tile inside the fused kernel.
// Global->LDS staging uses GLOBAL_LOAD_ASYNC_TO_LDS_B128 (ASYNCcnt) when the
// toolchain declares the builtin; otherwise falls back to float4 staging.
// ---------------------------------------------------------------------------

typedef float v2f __attribute__((ext_vector_type(2)));
typedef float v8f __attribute__((ext_vector_type(8)));
typedef int   v4i __attribute__((vector_size(16)));   // async builtin pointee

#define CDIM 256
#define ODIM 256
#define PDIM 4096   // h*w
#define NBATCH 4
#define PT 32       // p-tile width of the fused kernel
#define XPS 260     // XpT row stride (floats): 8B-aligned, bank-conflict-free

#if defined(__AMDGCN__) && __has_builtin(__builtin_amdgcn_global_load_async_to_lds_b128)
#define ASYNC_LDS 1
#else
#define ASYNC_LDS 0
#endif

__device__ __forceinline__ v8f wmma_f32(v2f a, v2f b, v8f c) {
    // V_WMMA_F32_16X16X4_F32 : D(16x16,f32) = A(16x4,f32) * B(4x16,f32) + C
    return __builtin_amdgcn_wmma_f32_16x16x4_f32(
        false, a, false, b, (short)0, c, false, false);
}

// copy 16 bytes global -> LDS (async on CDNA5 when available)
__device__ __forceinline__ void cp16(float* lds, const float* gsrc) {
#if ASYNC_LDS
    __builtin_amdgcn_global_load_async_to_lds_b128(
        (v4i*)gsrc, (v4i*)lds, 0, 0);
#else
    *(float4*)lds = *(const float4*)gsrc;
#endif
}

__device__ __forceinline__ void wait_async() {
#if ASYNC_LDS
#if __has_builtin(__builtin_amdgcn_s_wait_asynccnt)
    __builtin_amdgcn_s_wait_asynccnt(0);
#else
    asm volatile("s_wait_asynccnt 0" ::: "memory");
#endif
#endif
}

// aligned 8-byte LDS fragment load (lowered to ds_load_b64)
__device__ __forceinline__ v2f ld2(const float* p) {
    return *(const v2f*)p;
}

// ---------------------------------------------------------------------------
// Kernel: per-pixel channel norms  norms[b,p] = sqrt(sum_c x[b,c,p]^2)
// ---------------------------------------------------------------------------
__global__ __launch_bounds__(256) void k_norms(const float* __restrict__ x,
                                               float* __restrict__ norms) {
    int idx = blockIdx.x * 256 + threadIdx.x;       // 0 .. 4*4096-1
    int b = idx >> 12;
    int p = idx & (PDIM - 1);
    const float* xp = x + (size_t)b * CDIM * PDIM + p;
    float s = 0.f;
#pragma unroll 8
    for (int c = 0; c < CDIM; ++c) {
        float v = xp[(size_t)c * PDIM];
        s += v * v;
    }
    norms[idx] = sqrtf(s);
}

// ---------------------------------------------------------------------------
// Kernel: tx[b,o,q] = sum_c W[o,c] x[b,c,q] + bias[o]
// grid = (PDIM/64, NBATCH), block = 256 (8 waves).
// Wave w owns o-tiles 2w, 2w+1 x all four 16-wide q-tiles (8 acc tiles).
// ---------------------------------------------------------------------------
__global__ __launch_bounds__(256) void k_tx(const float* __restrict__ x,
                                            const float* __restrict__ W,
                                            const float* __restrict__ bias,
                                            float* __restrict__ tx) {
    const int b  = blockIdx.y;
    const int q0 = blockIdx.x * 64;
    const int t    = threadIdx.x;
    const int wv   = t >> 5;
    const int lane = t & 31;
    const int mr   = lane & 15;
    const int khi  = (lane < 16) ? 0 : 2;   // K-half select for A/B fragments
    const int mofs = (lane < 16) ? 0 : 8;   // M-half select for C/D tiles

    __shared__ float Wl[256][36];   // 256 x 32 slab of W (stride 36: 16B rows)
    __shared__ float Xl[32][72];    // 32 x 64 slab of X  (stride 72: 16B rows)

    v8f acc[8];
#pragma unroll
    for (int i = 0; i < 8; ++i) acc[i] = (v8f)0.f;

    const float* xb = x + (size_t)b * CDIM * PDIM;

    for (int c0 = 0; c0 < CDIM; c0 += 32) {
        __syncthreads();
        {   // stage W[*, c0:c0+32] -> LDS (256 rows x 32)
            int rr = t >> 3;             // 0..31
            int cc = (t & 7) * 4;        // 0..28
#pragma unroll
            for (int ps = 0; ps < 8; ++ps) {
                int row = ps * 32 + rr;
                cp16(&Wl[row][cc], W + (size_t)row * CDIM + c0 + cc);
            }
        }
        {   // stage X[c0:c0+32, q0:q0+64] -> LDS
            int rr = t >> 3;             // 0..31
            int cc = (t & 7) * 8;        // 0..56
            const float* src = xb + (size_t)(c0 + rr) * PDIM + q0 + cc;
            cp16(&Xl[rr][cc], src);
            cp16(&Xl[rr][cc + 4], src + 4);
        }
        wait_async();
        __syncthreads();

#pragma unroll
        for (int kk = 0; kk < 32; kk += 4) {
            v2f afr[2];
#pragma unroll
            for (int j = 0; j < 2; ++j) {
                int orow = (2 * wv + j) * 16 + mr;
                afr[j] = ld2(&Wl[orow][kk + khi]);       // contiguous -> b64
            }
            v2f bfr[4];
#pragma unroll
            for (int n = 0; n < 4; ++n) {
                bfr[n].x = Xl[kk + khi][n * 16 + mr];
                bfr[n].y = Xl[kk + khi + 1][n * 16 + mr];
            }
#pragma unroll
            for (int j = 0; j < 2; ++j)
#pragma unroll
                for (int n = 0; n < 4; ++n)
                    acc[j * 4 + n] = wmma_f32(afr[j], bfr[n], acc[j * 4 + n]);
        }
    }

    // store tx (+bias). C/D layout: VGPR r -> M = r + mofs, N = mr.
#pragma unroll
    for (int j = 0; j < 2; ++j) {
        int obase = (2 * wv + j) * 16;
#pragma unroll
        for (int n = 0; n < 4; ++n) {
            int q = q0 + n * 16 + mr;
#pragma unroll
            for (int r = 0; r < 8; ++r) {
                int o = obase + r + mofs;
                tx[((size_t)b * ODIM + o) * PDIM + q] = acc[j * 4 + n][r] + bias[o];
            }
        }
    }
}

// ---------------------------------------------------------------------------
// Fused kernel: per (batch, 32-wide p-tile)
//   stage 1: G[p_tile, q0:q0+128] = Xp^T Xq  (K=256 via 32-row LDS slabs);
//            wave w owns q-subtile w and both 16-wide p-subtiles (B reused 2x).
//            Xp kept transposed (XpT[p][c]) so A fragments are single b64s.
//            sim = relu(G / max(n_p n_q, eps))^2  -> LDS strip 32x128
//   stage 2: C[256, p_tile] += tx[:, q-chunk] . sim^T  (K=128 via 32-col
//            slabs); wave w owns o-tiles 2w,2w+1 x both p-subtiles
//            (A and B fragments each reused 2x, all contiguous b64 loads)
// grid = (PDIM/32, NBATCH), block = 256 (8 waves).
// ---------------------------------------------------------------------------
__global__ __launch_bounds__(256) void k_fused(const float* __restrict__ x,
                                               const float* __restrict__ tx,
                                               const float* __restrict__ norms,
                                               float* __restrict__ out) {
    const int b  = blockIdx.y;
    const int p0 = blockIdx.x * PT;
    const int t    = threadIdx.x;
    const int wv   = t >> 5;
    const int lane = t & 31;
    const int mr   = lane & 15;
    const int khi  = (lane < 16) ? 0 : 2;
    const int mofs = (lane < 16) ? 0 : 8;

    __shared__ float XpT[PT][XPS];      // resident X[:, p0:p0+32], TRANSPOSED
    __shared__ float simS[PT][132];     // sim strip, padded
    __shared__ float slab[256 * 36];    // union: Xq [32][136] / tx [256][36]
    __shared__ float npS[PT];
    __shared__ float nqS[128];

    const float* xb  = x  + (size_t)b * CDIM * PDIM;
    const float* txb = tx + (size_t)b * ODIM * PDIM;
    const float* nb  = norms + (size_t)b * PDIM;

    // load resident p-column block of X, transposing into XpT[p][c].
    // Global side stays coalesced (consecutive threads -> consecutive p).
    for (int i = t; i < CDIM * PT; i += 256) {
        int c = i >> 5;          // 0..255
        int p = i & (PT - 1);    // 0..31
        XpT[p][c] = xb[(size_t)c * PDIM + p0 + p];
    }
    if (t < PT) npS[t] = nb[p0 + t];

    v8f cacc[4];
#pragma unroll
    for (int i = 0; i < 4; ++i) cacc[i] = (v8f)0.f;

    for (int q0 = 0; q0 < PDIM; q0 += 128) {
        if (t < 128) nqS[t] = nb[q0 + t];

        // ---- stage 1: Gram strip; wave wv owns q-subtile [wv*16, wv*16+16)
        v8f g[2];
        g[0] = (v8f)0.f; g[1] = (v8f)0.f;
        for (int c0 = 0; c0 < CDIM; c0 += 32) {
            __syncthreads();            // slab free
            {   // stage X[c0:c0+32, q0:q0+128] -> slab (stride 136)
                int rr = t >> 3;        // 0..31
                int cc = (t & 7) * 16;  // 0..112
                const float* src = xb + (size_t)(c0 + rr) * PDIM + q0 + cc;
                float* dst = &slab[rr * 136 + cc];
#pragma unroll
                for (int v4 = 0; v4 < 4; ++v4)
                    cp16(dst + v4 * 4, src + v4 * 4);
            }
            wait_async();
            __syncthreads();            // slab visible (and XpT on first pass)
#pragma unroll
            for (int kk = 0; kk < 32; kk += 4) {
                v2f bfq;  // B[k=c][n=q] = Xq[c][q]
                bfq.x = slab[(kk + khi) * 136 + wv * 16 + mr];
                bfq.y = slab[(kk + khi + 1) * 136 + wv * 16 + mr];
#pragma unroll
                for (int pj = 0; pj < 2; ++pj) {
                    // A[m=p][k=c] = XpT[p][c] : contiguous -> single b64
                    v2f a = ld2(&XpT[pj * 16 + mr][c0 + kk + khi]);
                    g[pj] = wmma_f32(a, bfq, g[pj]);
                }
            }
        }

        // ---- transform to sim and park in LDS
        {
            float nq = nqS[wv * 16 + mr];
#pragma unroll
            for (int pj = 0; pj < 2; ++pj) {
#pragma unroll
                for (int r = 0; r < 8; ++r) {
                    float np = npS[pj * 16 + r + mofs];
                    float d  = fmaxf(np * nq, 1e-8f);
                    float v  = g[pj][r] / d;
                    v = fmaxf(v, 0.f);
                    simS[pj * 16 + r + mofs][wv * 16 + mr] = v * v;
                }
            }
        }
        __syncthreads();                // sim visible; slab free

        // ---- stage 2: C[o, p_tile] += tx[:, q-chunk] . sim^T
        for (int kq = 0; kq < 128; kq += 32) {
            {   // stage tx[0:256, q0+kq : +32] -> slab (stride 36)
                int rr = t >> 3;        // 0..31
                int cc = (t & 7) * 4;   // 0..28
#pragma unroll
                for (int ps = 0; ps < 8; ++ps) {
                    int row = ps * 32 + rr;
                    cp16(&slab[row * 36 + cc],
                         txb + (size_t)row * PDIM + q0 + kq + cc);
                }
            }
            wait_async();
            __syncthreads();            // slab visible
#pragma unroll
            for (int kk = 0; kk < 32; kk += 4) {
                v2f bf[2];  // B[k=q][n=p] = sim[p][q] : contiguous -> b64
#pragma unroll
                for (int pj = 0; pj < 2; ++pj)
                    bf[pj] = ld2(&simS[pj * 16 + mr][kq + kk + khi]);
#pragma unroll
                for (int j = 0; j < 2; ++j) {
                    int orow = (wv * 2 + j) * 16 + mr;
                    // A[m=o][k=q] = tx[o][q] : contiguous -> b64
                    v2f a = ld2(&slab[orow * 36 + kk + khi]);
#pragma unroll
                    for (int pj = 0; pj < 2; ++pj)
                        cacc[j * 2 + pj] = wmma_f32(a, bf[pj], cacc[j * 2 + pj]);
                }
            }
            __syncthreads();            // slab free for next K-slab
        }
    }

    // ---- store out[b, o, p0:p0+32]
#pragma unroll
    for (int j = 0; j < 2; ++j) {
#pragma unroll
        for (int pj = 0; pj < 2; ++pj) {
#pragma unroll
            for (int r = 0; r < 8; ++r) {
                int o = (wv * 2 + j) * 16 + r + mofs;
                out[((size_t)b * ODIM + o) * PDIM + p0 + pj * 16 + mr] =
                    cacc[j * 2 + pj][r];
            }
        }
    }
}

// ---------------------------------------------------------------------------
extern "C" void kernel_launch(void* const* d_in, const int* in_sizes, int n_in,
                              void* d_out, int out_size, void* d_ws, size_t ws_size,
                              hipStream_t stream) {
    const float* x    = (const float*)d_in[0];   // (4,256,64,64)
    const float* W    = (const float*)d_in[1];   // (256,256)
    const float* bias = (const float*)d_in[2];   // (256,)
    float* out = (float*)d_out;                  // (4,256,64,64)

    float* tx    = (float*)d_ws;                       // 4*256*4096 floats
    float* norms = tx + (size_t)NBATCH * ODIM * PDIM;  // 4*4096 floats

    k_norms<<<dim3((NBATCH * PDIM) / 256), dim3(256), 0, stream>>>(x, norms);

    k_tx<<<dim3(PDIM / 64, NBATCH), dim3(256), 0, stream>>>(x, W, bias, tx);

    k_fused<<<dim3(PDIM / PT, NBATCH), dim3(256), 0, stream>>>(x, tx, norms, out);
}